// TBIP_31318901522613
// MI455X (gfx1250) — compile-verified
//
#include <hip/hip_runtime.h>

// ---------------------------------------------------------------------------
// TBIP rate kernel for MI455X (gfx1250, wave32).
// rate[b,v] = w[b] * sum_k docc[b,k] * exp2( P[k,v] + ipl[b]*Q[k,v] )
//   P[k,v]  = (obj_loc + obj_scale*obj_eps) * log2(e)   (obj exp folded in!)
//   Q[k,v]  =  ideo_loc + ideo_scale*ideo_eps
//   docc    =  exp(doc_loc + doc_scale*doc_eps)[gathered rows]
//   ipl[b]  = (ip_loc + ip_scale*ip_eps)[gathered] * log2(e)
// Inner loop is the provable minimum: 2x v_fma_f32 + 1x v_exp_f32 per element.
// DS traffic in the hot loop: 1x ds_load_b64 (P,Q) + 2x ds_load_b128
// (wave-uniform docc broadcasts) per k iteration.
// ---------------------------------------------------------------------------

namespace {
constexpr int kD = 50000;
constexpr int kK = 50;
constexpr int kV = 10000;
constexpr int kA = 512;
constexpr int kB = 128;

constexpr int VT = 32;               // v-tile per block (== wave32 width)
constexpr int BT = 64;               // b-tile per block
constexpr int NT = 256;              // threads per block (8 waves)
constexpr int BPT = BT / (NT / VT);  // 8 b's per thread
constexpr int NPQ = (kK * VT + NT - 1) / NT;  // P/Q elements per thread (7)
constexpr float LOG2E = 1.4426950408889634f;
}  // namespace

// gfx1250 async global->LDS path (device pass only; host pass gets fallback).
#if defined(__gfx1250__) && __has_builtin(__builtin_amdgcn_global_load_async_to_lds_b128)
#define TBIP_ASYNC 1
#else
#define TBIP_ASYNC 0
#endif

#if TBIP_ASYNC
typedef int tbip_v4i __attribute__((vector_size(16)));
typedef __attribute__((address_space(1))) tbip_v4i* tbip_g4p;  // global int4*
typedef __attribute__((address_space(3))) tbip_v4i* tbip_l4p;  // LDS int4*
#endif

__device__ inline void tbip_wait_async_then_sync() {
#if TBIP_ASYNC
#if __has_builtin(__builtin_amdgcn_s_wait_asynccnt)
  __builtin_amdgcn_s_wait_asynccnt(0);
#else
  asm volatile("s_wait_asynccnt 0" ::: "memory");
#endif
#endif
  __syncthreads();
}

// Stage one (K x VT) tile: g[k*kV + vbase + x] -> lds[k*VT + x], 16B chunks.
__device__ inline void tbip_stage_tile(const float* __restrict__ g,
                                       float* __restrict__ lds,
                                       int vbase, int tid) {
  for (int i = tid; i < kK * (VT / 4); i += NT) {
    const int k = i / (VT / 4);
    const int x = (i % (VT / 4)) * 4;
    const int v0 = vbase + x;
    if (v0 + 4 <= kV) {
      const float* gp = g + (size_t)k * kV + v0;
      float* lp = lds + k * VT + x;
#if TBIP_ASYNC
      // Generic->AS1 / AS3 via integer round-trip (low 32 bits == LDS offset).
      tbip_g4p gp1 = (tbip_g4p)(unsigned long long)gp;
      tbip_l4p lp3 = (tbip_l4p)(unsigned int)(unsigned long long)lp;
      __builtin_amdgcn_global_load_async_to_lds_b128(gp1, lp3, 0, 0);
#else
      *(float4*)lp = *(const float4*)gp;
#endif
    }
  }
}

__global__ __launch_bounds__(NT) void tbip_rate_kernel(
    const float* __restrict__ doc_loc, const float* __restrict__ doc_scale,
    const float* __restrict__ doc_eps,
    const float* __restrict__ obj_loc, const float* __restrict__ obj_scale,
    const float* __restrict__ obj_eps,
    const float* __restrict__ ideo_loc, const float* __restrict__ ideo_scale,
    const float* __restrict__ ideo_eps,
    const float* __restrict__ ip_loc, const float* __restrict__ ip_scale,
    const float* __restrict__ ip_eps,
    const float* __restrict__ author_weights,
    const int* __restrict__ document_indices,
    const int* __restrict__ author_indices,
    float* __restrict__ out) {
  __shared__ float s_raw[6 * kK * VT];  // 38.4 KB staged params
  __shared__ float s_docc[kK * BT];     // 12.8 KB, transposed [k][bl]
  __shared__ float s_ipl[BT];
  __shared__ float s_w[BT];

  const int tid = threadIdx.x;
  const int vbase = blockIdx.x * VT;
  const int bbase = blockIdx.y * BT;

  float* t_ol = s_raw + 0 * kK * VT;
  float* t_os = s_raw + 1 * kK * VT;
  float* t_oe = s_raw + 2 * kK * VT;
  float* t_il = s_raw + 3 * kK * VT;
  float* t_is = s_raw + 4 * kK * VT;
  float* t_ie = s_raw + 5 * kK * VT;
  float* s_PQ = s_raw;  // interleaved {P,Q} pairs, reuses tiles 0-1 after sync

  // Phase 0: kick off async staging of the six obj/ideo param tiles.
  tbip_stage_tile(obj_loc, t_ol, vbase, tid);
  tbip_stage_tile(obj_scale, t_os, vbase, tid);
  tbip_stage_tile(obj_eps, t_oe, vbase, tid);
  tbip_stage_tile(ideo_loc, t_il, vbase, tid);
  tbip_stage_tile(ideo_scale, t_is, vbase, tid);
  tbip_stage_tile(ideo_eps, t_ie, vbase, tid);

  // Phase 1 (overlaps the async copies): gathered per-b terms.
  for (int i = tid; i < BT * kK; i += NT) {
    const int bl = i / kK, k = i % kK;
    const int d = document_indices[bbase + bl];
    const size_t off = (size_t)d * kK + k;
    const float x = __builtin_fmaf(doc_scale[off], doc_eps[off], doc_loc[off]);
    s_docc[k * BT + bl] = exp2f(x * LOG2E);  // exp(x) as native v_exp_f32
  }
  if (tid < BT) {
    const int a = author_indices[bbase + tid];
    s_ipl[tid] = __builtin_fmaf(ip_scale[a], ip_eps[a], ip_loc[a]) * LOG2E;
    s_w[tid] = author_weights[a];
  }

  tbip_wait_async_then_sync();

  // Phase 2: fold obj into the exponent (P) and build Q — no transcendentals.
  // Read everything to registers, barrier, then overwrite tiles 0-1 with the
  // interleaved {P,Q} layout (cross-thread alias, hence the split barriers).
  float Pr[NPQ], Qr[NPQ];
#pragma unroll
  for (int t = 0; t < NPQ; ++t) {
    const int i = tid + t * NT;
    float P = -1.0e30f, Q = 0.0f;  // exp2 -> 0 for out-of-range v lanes
    if (i < kK * VT) {
      const int vo = i % VT;
      if (vbase + vo < kV) {
        P = __builtin_fmaf(t_os[i], t_oe[i], t_ol[i]) * LOG2E;
        Q = __builtin_fmaf(t_is[i], t_ie[i], t_il[i]);
      }
    }
    Pr[t] = P;
    Qr[t] = Q;
  }
  __syncthreads();  // all raw reads complete before aliasing writes
#pragma unroll
  for (int t = 0; t < NPQ; ++t) {
    const int i = tid + t * NT;
    if (i < kK * VT) {
      s_PQ[2 * i + 0] = Pr[t];
      s_PQ[2 * i + 1] = Qr[t];
    }
  }
  __syncthreads();

  // Phase 3: trans-bound main loop. Lane = one v (coalesced / conflict-free),
  // 8 b's per thread; docc/ipl reads are wave-uniform LDS broadcasts.
  const int vo = tid % VT;
  const int bl0 = (tid / VT) * BPT;

  float acc[BPT], ipl[BPT];
#pragma unroll
  for (int j = 0; j < BPT; ++j) {
    acc[j] = 0.0f;
    ipl[j] = s_ipl[bl0 + j];
  }

  for (int k = 0; k < kK; ++k) {
    const float2 pq = *(const float2*)&s_PQ[2 * (k * VT + vo)];  // ds_load_b64
    const float4 d0 = *(const float4*)&s_docc[k * BT + bl0];      // ds_load_b128
    const float4 d1 = *(const float4*)&s_docc[k * BT + bl0 + 4];  // (broadcast)
    const float dcf[BPT] = {d0.x, d0.y, d0.z, d0.w, d1.x, d1.y, d1.z, d1.w};
#pragma unroll
    for (int j = 0; j < BPT; ++j) {
      const float e = exp2f(__builtin_fmaf(ipl[j], pq.y, pq.x));  // fma + exp
      acc[j] = __builtin_fmaf(dcf[j], e, acc[j]);                 // fma
    }
  }

  const int v = vbase + vo;
  if (v < kV) {
#pragma unroll
    for (int j = 0; j < BPT; ++j) {
      out[(size_t)(bbase + bl0 + j) * kV + v] = s_w[bl0 + j] * acc[j];
    }
  }
}

extern "C" void kernel_launch(void* const* d_in, const int* in_sizes, int n_in,
                              void* d_out, int out_size, void* d_ws, size_t ws_size,
                              hipStream_t stream) {
  (void)in_sizes; (void)n_in; (void)out_size; (void)d_ws; (void)ws_size;
  const float* doc_loc = (const float*)d_in[0];
  const float* doc_scale = (const float*)d_in[1];
  const float* doc_eps = (const float*)d_in[2];
  const float* obj_loc = (const float*)d_in[3];
  const float* obj_scale = (const float*)d_in[4];
  const float* obj_eps = (const float*)d_in[5];
  const float* ideo_loc = (const float*)d_in[6];
  const float* ideo_scale = (const float*)d_in[7];
  const float* ideo_eps = (const float*)d_in[8];
  const float* ip_loc = (const float*)d_in[9];
  const float* ip_scale = (const float*)d_in[10];
  const float* ip_eps = (const float*)d_in[11];
  const float* author_weights = (const float*)d_in[12];
  const int* document_indices = (const int*)d_in[13];
  const int* author_indices = (const int*)d_in[14];
  float* out = (float*)d_out;

  dim3 grid((kV + VT - 1) / VT, kB / BT, 1);  // 313 x 2 blocks
  tbip_rate_kernel<<<grid, dim3(NT, 1, 1), 0, stream>>>(
      doc_loc, doc_scale, doc_eps, obj_loc, obj_scale, obj_eps, ideo_loc,
      ideo_scale, ideo_eps, ip_loc, ip_scale, ip_eps, author_weights,
      document_indices, author_indices, out);
}